// TopKSparseAutoencoder_72653666779437
// MI455X (gfx1250) — compile-verified
//
#include <hip/hip_runtime.h>

typedef __attribute__((ext_vector_type(2))) float v2f;
typedef __attribute__((ext_vector_type(8))) float v8f;

#define B_DIM 2048
#define I_DIM 768
#define H_DIM 24576
#define K_SEL 50

// ---- encoder GEMM tiling ----
#define MT 64          // batch rows per block (4 sub-tiles of 16 per wave)
#define NT 128         // hidden cols per block (8 waves x 16)
#define KC 64          // K chunk staged in LDS
#define LDA 68         // padded LDS row stride in dwords: 16B-aligned rows, conflict-free
#define NCHUNK (I_DIM / KC)   // 12

// rank-1 bias correction: corr[n] = sum_i pre_bias[i] * W_enc[n,i]
__global__ __launch_bounds__(256)
void sae_enc_bias_corr(const float* __restrict__ W_enc,
                       const float* __restrict__ pre_bias,
                       float* __restrict__ corr)
{
    const int n    = blockIdx.x * 8 + (threadIdx.x >> 5);
    const int lane = threadIdx.x & 31;
    float s = 0.0f;
    for (int i = lane; i < I_DIM; i += 32)
        s += W_enc[(size_t)n * I_DIM + i] * pre_bias[i];
    #pragma unroll
    for (int off = 16; off > 0; off >>= 1)
        s += __shfl_down(s, off, 32);
    if (lane == 0) corr[n] = s;
}

__global__ __launch_bounds__(256)
void sae_encode_wmma(const float* __restrict__ x,
                     const float* __restrict__ W_enc,
                     const float* __restrict__ corr,
                     const float* __restrict__ latent_bias,
                     float* __restrict__ pre_acts,
                     float* __restrict__ sparse_code)
{
    __shared__ __attribute__((aligned(16))) float As[2][MT * LDA];  // x tile [m][k]
    __shared__ __attribute__((aligned(16))) float Ws[2][NT * LDA];  // W tile [n][k]

    const int n0   = blockIdx.x * NT;
    const int m0   = blockIdx.y * MT;
    const int tid  = threadIdx.x;
    const int wave = tid >> 5;
    const int lane = tid & 31;
    const int half = lane >> 4;      // 0: K0/K1 lanes, 1: K2/K3 lanes
    const int lq   = lane & 15;

    // 32-bit LDS byte offsets for async-to-LDS (low 32 bits of flat shared addr)
    const unsigned asBase = (unsigned)(unsigned long long)(&As[0][0]);
    const unsigned wsBase = (unsigned)(unsigned long long)(&Ws[0][0]);

    // issue one K-chunk of async B128 copies (12 instructions/thread)
    auto issue_chunk = [&](int kc, int buf) {
        // A tile: 64 rows x 64 cols = 1024 x b128
        #pragma unroll
        for (int j = 0; j < 4; ++j) {
            int idx = j * 256 + tid;
            int r = idx >> 4, c4 = (idx & 15) << 2;
            unsigned lds = asBase + (unsigned)((buf * (MT * LDA) + r * LDA + c4) << 2);
            unsigned long long ga =
                (unsigned long long)(x + (size_t)(m0 + r) * I_DIM + kc + c4);
            asm volatile("global_load_async_to_lds_b128 %0, %1, off"
                         :: "v"(lds), "v"(ga) : "memory");
        }
        // W tile: 128 rows x 64 cols = 2048 x b128
        #pragma unroll
        for (int j = 0; j < 8; ++j) {
            int idx = j * 256 + tid;
            int r = idx >> 4, c4 = (idx & 15) << 2;
            unsigned lds = wsBase + (unsigned)((buf * (NT * LDA) + r * LDA + c4) << 2);
            unsigned long long ga =
                (unsigned long long)(W_enc + (size_t)(n0 + r) * I_DIM + kc + c4);
            asm volatile("global_load_async_to_lds_b128 %0, %1, off"
                         :: "v"(lds), "v"(ga) : "memory");
        }
    };

    v8f acc[4] = {};

    issue_chunk(0, 0);
    for (int ch = 0; ch < NCHUNK; ++ch) {
        const int cur = ch & 1;
        if (ch + 1 < NCHUNK) {
            issue_chunk((ch + 1) * KC, cur ^ 1);       // prefetch next chunk
            asm volatile("s_wait_asynccnt 0xc" ::: "memory");  // chunk ch landed
        } else {
            asm volatile("s_wait_asynccnt 0x0" ::: "memory");
        }
        __syncthreads();

        #pragma unroll
        for (int kk = 0; kk < KC; kk += 4) {
            // B fragment: B[k][n] = Ws[n][k]; lanes 0-15 -> K0/K1, 16-31 -> K2/K3
            v2f b;
            b.x = Ws[cur][(wave * 16 + lq) * LDA + kk + 2 * half];
            b.y = Ws[cur][(wave * 16 + lq) * LDA + kk + 2 * half + 1];
            #pragma unroll
            for (int mt = 0; mt < 4; ++mt) {
                v2f a;
                a.x = As[cur][(mt * 16 + lq) * LDA + kk + 2 * half];
                a.y = As[cur][(mt * 16 + lq) * LDA + kk + 2 * half + 1];
                acc[mt] = __builtin_amdgcn_wmma_f32_16x16x4_f32(
                    false, a, false, b, (short)0, acc[mt], false, false);
            }
        }
        __syncthreads();
    }

    // epilogue: C/D layout -> lanes 0-15: M=j, N=lane; lanes 16-31: M=j+8, N=lane-16
    const int ncol = n0 + wave * 16 + lq;
    const float colAdd = latent_bias[ncol] - corr[ncol];
    #pragma unroll
    for (int mt = 0; mt < 4; ++mt) {
        #pragma unroll
        for (int j = 0; j < 8; ++j) {
            int m = mt * 16 + j + 8 * half;
            size_t off = (size_t)(m0 + m) * H_DIM + ncol;
            pre_acts[off]    = acc[mt][j] + colAdd;
            sparse_code[off] = 0.0f;     // zero-init; topk kernel scatters nnz later
        }
    }
}

// ---- top-K selection + sparse decode ----
#define TPB   512
#define PER_T (H_DIM / TPB)   // 48 values per thread, register-resident

__global__ __launch_bounds__(TPB)
void sae_topk_decode(const float* __restrict__ pre_acts,
                     const float* __restrict__ W_dec,
                     const float* __restrict__ pre_bias,
                     float* __restrict__ sparse_code,
                     float* __restrict__ recon,
                     float* __restrict__ topk_vals,
                     float* __restrict__ topk_idx)
{
    __shared__ float s_val[K_SEL];
    __shared__ int   s_idx[K_SEL];
    __shared__ float r_val[TPB / 32];
    __shared__ int   r_idx[TPB / 32];

    const int row  = blockIdx.x;
    const int tid  = threadIdx.x;
    const int wave = tid >> 5;
    const int lane = tid & 31;

    // load + relu, one coalesced pass; row stays in registers for all 50 passes
    float v[PER_T];
    #pragma unroll
    for (int t = 0; t < PER_T; ++t) {
        float p = pre_acts[(size_t)row * H_DIM + t * TPB + tid];
        v[t] = p > 0.0f ? p : 0.0f;
    }

    // iterative selection: max over keys strictly below previous (value desc, idx asc)
    float prev_v = 3.402823466e38f;
    int   prev_i = -1;

    for (int s = 0; s < K_SEL; ++s) {
        float bv = -1.0f;          // relu values are >= 0
        int   bi = 0x7fffffff;
        #pragma unroll
        for (int t = 0; t < PER_T; ++t) {
            int   h   = t * TPB + tid;
            float val = v[t];
            bool elig = (val < prev_v) || (val == prev_v && h > prev_i);
            if (elig && (val > bv || (val == bv && h < bi))) { bv = val; bi = h; }
        }
        // intra-wave argmax reduction (wave32)
        #pragma unroll
        for (int off = 16; off > 0; off >>= 1) {
            float ov = __shfl_down(bv, off, 32);
            int   oi = __shfl_down(bi, off, 32);
            if (ov > bv || (ov == bv && oi < bi)) { bv = ov; bi = oi; }
        }
        if (lane == 0) { r_val[wave] = bv; r_idx[wave] = bi; }
        __syncthreads();
        if (tid == 0) {
            float gv = r_val[0]; int gi = r_idx[0];
            for (int w = 1; w < TPB / 32; ++w) {
                float ov = r_val[w]; int oi = r_idx[w];
                if (ov > gv || (ov == gv && oi < gi)) { gv = ov; gi = oi; }
            }
            s_val[s] = gv; s_idx[s] = gi;
        }
        __syncthreads();
        prev_v = s_val[s];
        prev_i = s_idx[s];
    }

    // emit top-k outputs and scatter into sparse_code
    if (tid < K_SEL) {
        topk_vals[(size_t)row * K_SEL + tid] = s_val[tid];
        topk_idx [(size_t)row * K_SEL + tid] = (float)s_idx[tid];
        sparse_code[(size_t)row * H_DIM + s_idx[tid]] = s_val[tid];
    }
    __syncthreads();

    // sparse decode: rec[row,i] = pre_bias[i] + sum_k val_k * W_dec[i, idx_k]
    for (int i = tid; i < I_DIM; i += TPB) {
        float accum = pre_bias[i];
        #pragma unroll 10
        for (int k = 0; k < K_SEL; ++k)
            accum += s_val[k] * W_dec[(size_t)i * H_DIM + s_idx[k]];
        recon[(size_t)row * I_DIM + i] = accum;
    }
}

extern "C" void kernel_launch(void* const* d_in, const int* in_sizes, int n_in,
                              void* d_out, int out_size, void* d_ws, size_t ws_size,
                              hipStream_t stream)
{
    const float* x           = (const float*)d_in[0];
    const float* W_enc       = (const float*)d_in[1];
    const float* W_dec       = (const float*)d_in[2];
    const float* pre_bias    = (const float*)d_in[3];
    const float* latent_bias = (const float*)d_in[4];

    // outputs concatenated flat in return order
    float* out         = (float*)d_out;
    float* recon       = out;                                    // [2048, 768]
    float* sparse_code = recon + (size_t)B_DIM * I_DIM;          // [2048, 24576]
    float* pre_acts    = sparse_code + (size_t)B_DIM * H_DIM;    // [2048, 24576]
    float* topk_vals   = pre_acts + (size_t)B_DIM * H_DIM;       // [2048, 50]
    float* topk_idx    = topk_vals + (size_t)B_DIM * K_SEL;      // [2048, 50]

    // scratch for the rank-1 bias correction (H floats). Prefer d_ws; if the
    // workspace is too small, borrow the topk_idx output region: it is only
    // written by the final kernel, after the encoder has consumed corr.
    float* corr = (ws_size >= (size_t)H_DIM * sizeof(float)) ? (float*)d_ws
                                                             : topk_idx;

    sae_enc_bias_corr<<<H_DIM / 8, 256, 0, stream>>>(W_enc, pre_bias, corr);

    dim3 grid1(H_DIM / NT, B_DIM / MT);   // (192, 32)
    sae_encode_wmma<<<grid1, 256, 0, stream>>>(x, W_enc, corr, latent_bias,
                                               pre_acts, sparse_code);
    sae_topk_decode<<<B_DIM, TPB, 0, stream>>>(pre_acts, W_dec, pre_bias,
                                               sparse_code, recon,
                                               topk_vals, topk_idx);
}